// Attention_5454608466763
// MI455X (gfx1250) — compile-verified
//
#include <hip/hip_runtime.h>

// ---------------------------------------------------------------------------
// MI455X / gfx1250 fp32 attention with V_WMMA_F32_16X16X4_F32 throughout.
// Shapes fixed by the reference: B=2, S=2048, D=1024, H=16, U=64.
// ---------------------------------------------------------------------------

typedef __attribute__((ext_vector_type(2))) float v2f;
typedef __attribute__((ext_vector_type(8))) float v8f;

#define BATCH  2
#define SEQ    2048
#define DMODEL 1024
#define NHEAD  16
#define HDIM   64
#define DENSE  1024   // NHEAD*HDIM

// D = A(16x4 f32) * B(4x16 f32) + C(16x16 f32)
// A lane layout: M = lane&15, K = 2*(lane>>4) + {0,1}  -> contiguous float2
// B lane layout: N = lane&15, K = 2*(lane>>4) + {0,1}
// C lane layout: vgpr j holds M = j + 8*(lane>>4), N = lane&15
static __device__ __forceinline__ v8f wmma_f32(v2f a, v2f b, v8f c) {
    return __builtin_amdgcn_wmma_f32_16x16x4_f32(
        /*neg_a=*/false, a, /*neg_b=*/false, b,
        /*c_mod=*/(short)0, c, /*reuse_a=*/false, /*reuse_b=*/false);
}

// ---------------------------------------------------------------------------
// Kernel 1: qk = X @ Wqk + bqk ; v = X @ Wv + bv   stored as (B,H,S,U) fp32.
// One 16x16 output tile per wave; 4 waves / block; 32768 tiles total.
// ---------------------------------------------------------------------------
__global__ __launch_bounds__(128) void proj_kernel(
    const float* __restrict__ X,
    const float* __restrict__ Wqk, const float* __restrict__ bqk,
    const float* __restrict__ Wv,  const float* __restrict__ bv,
    float* __restrict__ qk_ws, float* __restrict__ v_ws)
{
    const int lane = threadIdx.x & 31;
    const int lo = lane & 15, hi = lane >> 4;
    const int wave = blockIdx.x * 4 + (threadIdx.x >> 5); // 0..32767
    const int mat  = wave >> 14;                          // 0: qk, 1: v
    const int t    = wave & 16383;
    const int m0   = (t >> 6) * 16;                       // 256 M-tiles
    const int n0   = (t & 63) * 16;                       // 64  N-tiles

    const float* W    = mat ? Wv : Wqk;
    const float* bias = mat ? bv : bqk;
    float*       dst  = mat ? v_ws : qk_ws;

    const float* arow = X + (size_t)(m0 + lo) * DMODEL + hi * 2;
    const float* bcol = W + (size_t)(hi * 2) * DENSE + n0 + lo;

    v8f acc = {};
    for (int k = 0; k < DMODEL; k += 4) {
        v2f a = *(const v2f*)(arow + k);
        v2f b;
        b.x = bcol[(size_t)k * DENSE];
        b.y = bcol[(size_t)(k + 1) * DENSE];
        acc = wmma_f32(a, b, acc);
    }

    const int ng = n0 + lo;
    const float bn = bias[ng];
    const int hh = ng >> 6, u = ng & (HDIM - 1);
    #pragma unroll
    for (int j = 0; j < 8; ++j) {
        int mg = m0 + j + 8 * hi;                  // row in flattened (B*S)
        int bb = mg >> 11, s = mg & (SEQ - 1);
        dst[((size_t)(bb * NHEAD + hh) * SEQ + s) * HDIM + u] = acc[j] + bn;
    }
}

// ---------------------------------------------------------------------------
// Kernel 2: per-(b,h) attention. One wave per 16-row q-tile.
//  Pass A: online softmax (running max/sum) over all 128 key tiles.
//  Pass B: recompute scores, emit final probs once, accumulate p @ v.
// ---------------------------------------------------------------------------
__global__ __launch_bounds__(128) void attn_kernel(
    const float* __restrict__ qk_ws, const float* __restrict__ v_ws,
    const int* __restrict__ mask,
    float* __restrict__ out_ctx, float* __restrict__ out_probs)
{
    __shared__ float rmaxS[4][16];
    __shared__ float rsumS[4][16];
    __shared__ float ptile[4][16][17];   // pad 17: conflict-free A-layout reads

    const int lane = threadIdx.x & 31;
    const int w    = threadIdx.x >> 5;
    const int lo = lane & 15, hi = lane >> 4;
    const int wave = blockIdx.x * 4 + w;     // 0..4095
    const int qt = wave & 127;               // q tile within sequence
    const int bh = wave >> 7;                // 0..31
    const int b  = bh >> 4, h = bh & 15;

    const float* qkb = qk_ws + (size_t)bh * SEQ * HDIM;
    const float* vb  = v_ws  + (size_t)bh * SEQ * HDIM;
    const int*   mb  = mask + b * SEQ;
    float* probsb = out_probs + (size_t)bh * SEQ * SEQ;

    // Preload A (the 16 q-rows of qk), A-layout: row lo, k = t*4 + 2*hi + {0,1}
    v2f areg[16];
    {
        const float* ap = qkb + (size_t)(qt * 16 + lo) * HDIM + hi * 2;
        #pragma unroll
        for (int t = 0; t < 16; ++t) areg[t] = *(const v2f*)(ap + t * 4);
    }

    float mrun[8], lsum[8];
    #pragma unroll
    for (int j = 0; j < 8; ++j) { mrun[j] = -1e30f; lsum[j] = 0.f; }

    // ---------------- Pass A: online max / sum ----------------
    for (int kt = 0; kt < 128; ++kt) {
        const int mk = mb[kt * 16 + lo];
        const float* bp = qkb + (size_t)(kt * 16 + lo) * HDIM + hi * 2;
        v8f acc = {};
        #pragma unroll
        for (int t = 0; t < 16; ++t) {
            v2f bv2 = *(const v2f*)(bp + t * 4);     // B: qk^T, also float2
            acc = wmma_f32(areg[t], bv2, acc);
        }
        #pragma unroll
        for (int j = 0; j < 8; ++j) {
            int qg = qt * 16 + j + 8 * hi;
            int kg = kt * 16 + lo;
            float s = acc[j] * 0.125f;               // / sqrt(64)
            if (!mk || qg == kg) s -= 10000.f;       // mask + no-self
            float tmax = s;                           // 16-lane row max
            tmax = fmaxf(tmax, __shfl_xor(tmax, 1));
            tmax = fmaxf(tmax, __shfl_xor(tmax, 2));
            tmax = fmaxf(tmax, __shfl_xor(tmax, 4));
            tmax = fmaxf(tmax, __shfl_xor(tmax, 8));
            float mnew  = fmaxf(mrun[j], tmax);
            float scale = __expf(mrun[j] - mnew);
            lsum[j] = lsum[j] * scale + __expf(s - mnew);
            mrun[j] = mnew;
        }
    }
    #pragma unroll
    for (int j = 0; j < 8; ++j) {
        float sum = lsum[j];
        sum += __shfl_xor(sum, 1);
        sum += __shfl_xor(sum, 2);
        sum += __shfl_xor(sum, 4);
        sum += __shfl_xor(sum, 8);
        if (lo == 0) { rmaxS[w][j + 8 * hi] = mrun[j]; rsumS[w][j + 8 * hi] = sum; }
    }
    // Broadcast per-row stats (same-wave LDS, hardware keeps DS ops in order)
    float cm[8], cis[8];
    #pragma unroll
    for (int j = 0; j < 8; ++j) {
        cm[j]  = rmaxS[w][j + 8 * hi];
        cis[j] = 1.0f / rsumS[w][j + 8 * hi];
    }

    // ---------------- Pass B: probs out + context = p @ v ----------------
    v8f ctx0 = {}, ctx1 = {}, ctx2 = {}, ctx3 = {};
    for (int kt = 0; kt < 128; ++kt) {
        const int mk = mb[kt * 16 + lo];
        const float* bp = qkb + (size_t)(kt * 16 + lo) * HDIM + hi * 2;
        v8f acc = {};
        #pragma unroll
        for (int t = 0; t < 16; ++t) {
            v2f bv2 = *(const v2f*)(bp + t * 4);
            acc = wmma_f32(areg[t], bv2, acc);
        }
        #pragma unroll
        for (int j = 0; j < 8; ++j) {
            int qg = qt * 16 + j + 8 * hi;
            int kg = kt * 16 + lo;
            float s = acc[j] * 0.125f;
            if (!mk || qg == kg) s -= 10000.f;
            float p = __expf(s - cm[j]) * cis[j];    // final normalized prob
            ptile[w][j + 8 * hi][lo] = p;            // stage C-layout -> LDS
            probsb[(size_t)qg * SEQ + kg] = p;       // single probs write
        }
        // context: A = ptile (read back in A-layout), B = v tile
        const float* vkb = vb + (size_t)(kt * 16 + hi * 2) * HDIM + lo;
        #pragma unroll
        for (int t = 0; t < 4; ++t) {
            v2f pa;
            pa.x = ptile[w][lo][t * 4 + hi * 2];
            pa.y = ptile[w][lo][t * 4 + hi * 2 + 1];
            const float* vk = vkb + (size_t)(t * 4) * HDIM;
            v2f b0, b1, b2, b3;
            b0.x = vk[0];  b0.y = vk[HDIM];
            b1.x = vk[16]; b1.y = vk[HDIM + 16];
            b2.x = vk[32]; b2.y = vk[HDIM + 32];
            b3.x = vk[48]; b3.y = vk[HDIM + 48];
            ctx0 = wmma_f32(pa, b0, ctx0);
            ctx1 = wmma_f32(pa, b1, ctx1);
            ctx2 = wmma_f32(pa, b2, ctx2);
            ctx3 = wmma_f32(pa, b3, ctx3);
        }
    }

    // context already normalized (p was final). Write (B,S,H*U).
    #pragma unroll
    for (int j = 0; j < 8; ++j) {
        int sg = qt * 16 + j + 8 * hi;
        float* o = out_ctx + ((size_t)(b * SEQ + sg)) * DENSE + h * HDIM + lo;
        o[0]  = ctx0[j];
        o[16] = ctx1[j];
        o[32] = ctx2[j];
        o[48] = ctx3[j];
    }
}

// ---------------------------------------------------------------------------
extern "C" void kernel_launch(void* const* d_in, const int* in_sizes, int n_in,
                              void* d_out, int out_size, void* d_ws, size_t ws_size,
                              hipStream_t stream) {
    (void)in_sizes; (void)n_in; (void)out_size; (void)ws_size;
    const float* X   = (const float*)d_in[0];
    const int*   msk = (const int*)d_in[1];
    const float* Wqk = (const float*)d_in[2];
    const float* bqk = (const float*)d_in[3];
    const float* Wv  = (const float*)d_in[4];
    const float* bv  = (const float*)d_in[5];

    float* out   = (float*)d_out;
    float* ctx   = out;                                    // B*S*DENSE
    float* probs = out + (size_t)BATCH * SEQ * DENSE;      // B*H*S*S

    float* qk_ws = (float*)d_ws;                           // 16 MB
    float* v_ws  = qk_ws + (size_t)BATCH * NHEAD * SEQ * HDIM; // +16 MB

    // 32768 projection tiles, 4 waves/block
    proj_kernel<<<8192, 128, 0, stream>>>(X, Wqk, bqk, Wv, bv, qk_ws, v_ws);
    // 4096 q-tiles, 4 waves/block
    attn_kernel<<<1024, 128, 0, stream>>>(qk_ws, v_ws, msk, ctx, probs);
}